// Lorentz_71365176590489
// MI455X (gfx1250) — compile-verified
//
#include <hip/hip_runtime.h>
#include <math.h>

// Lorentz (hyperboloid) softmax loss, MI455X / gfx1250.
//
// Memory-bound gather kernel: 16384*51 random 128-byte row gathers (~107 MB)
// from a 128 MB table that fits in the 192 MB L2 -> ~4.6 us HBM floor, less
// from L2. Compute (26M MACs + 800K transcendentals) is free, so the 32-dim
// Lorentz dot products are done on the matrix pipe with the only f32-accurate
// WMMA shape, V_WMMA_F32_16X16X4_F32 (arcosh near 1 forbids low precision).
//
// One wave (32 lanes, wave32) per batch element b:
//   A (16x4 f32)  = anchor row ui' (time dim negated) broadcast to all 16 rows,
//                   built from scalar-loaded values with v_cndmask (no vmem).
//   B (4x16 f32)  = 16 gathered neighbor rows per column-tile; K=32 contracted
//                   by chaining 8 WMMAs into the f32 accumulator.
//   D row-broadcast => every lane holds the dot for its column n directly;
//   only the final log-sum-exp needs a 5-step butterfly reduction.

typedef __attribute__((ext_vector_type(2))) float v2f;
typedef __attribute__((ext_vector_type(8))) float v8f;

#define DIM 32
#define NKS 50
#define WAVES_PER_BLOCK 8

__global__ __launch_bounds__(256) void lorentz_loss_kernel(
    const float* __restrict__ table,
    const int*   __restrict__ I,
    const int*   __restrict__ Ks,
    float*       __restrict__ out,
    int Bcount)
{
    const int lane = threadIdx.x & 31;
    // Force wave id into an SGPR so b / I[b] / anchor-row loads scalarize.
    const int wave = __builtin_amdgcn_readfirstlane((int)(threadIdx.x >> 5));
    const int b    = blockIdx.x * WAVES_PER_BLOCK + wave;
    if (b >= Bcount) return;   // wave-uniform: EXEC stays all-ones past here

    // ---- anchor row (wave-uniform -> s_load), negate time component -------
    const int ib = I[b];
    const float* __restrict__ ui = table + (size_t)ib * DIM;
    float u[DIM];
#pragma unroll
    for (int k = 0; k < DIM; ++k) u[k] = ui[k];
    u[0] = -u[0];   // Lorentz signature: lsp = sum(spatial) - time

    // ---- A fragments (16x4 f32 layout, ISA 7.12.2) ------------------------
    // VGPR0: K = 4c   (lanes 0-15) | 4c+2 (lanes 16-31)
    // VGPR1: K = 4c+1 (lanes 0-15) | 4c+3 (lanes 16-31)
    // All 16 M-rows identical (broadcast anchor): pure lane-half selects.
    const bool lowHalf = (lane < 16);
    v2f A[8];
#pragma unroll
    for (int c = 0; c < 8; ++c) {
        A[c].x = lowHalf ? u[4 * c]     : u[4 * c + 2];
        A[c].y = lowHalf ? u[4 * c + 1] : u[4 * c + 3];
    }

    const int hi   = lowHalf ? 0 : 2;     // lane-half K offset for B frags
    const int nlow = lane & 15;           // column within tile

    float d0     = 0.0f;   // -arcosh of the positive (n==0), held by lane 0
    float sumExp = 0.0f;

#pragma unroll
    for (int t = 0; t < 4; ++t) {          // 4 column tiles cover 50 (pad 64)
        const int  n     = t * 16 + nlow;
        const bool valid = (n < NKS);
        // lanes l and l+16 read the same neighbor row (column duplicated)
        const int  row   = valid ? Ks[b * NKS + n] : 0;
        const float* __restrict__ r = table + (size_t)row * DIM;

        // B fragments (4x16 f32, row-striped, mirrors A's K split)
        v2f Bv[8];
#pragma unroll
        for (int c = 0; c < 8; ++c) {
            Bv[c].x = r[4 * c + hi];
            Bv[c].y = r[4 * c + 1 + hi];
        }

        // Contract K=32 as 8 chained f32 WMMAs (matrix pipe, RNE, denorms ok)
        v8f acc = {};
#pragma unroll
        for (int c = 0; c < 8; ++c) {
            acc = __builtin_amdgcn_wmma_f32_16x16x4_f32(
                /*neg_a=*/false, A[c],
                /*neg_b=*/false, Bv[c],
                /*c_mod=*/(short)0, acc,
                /*reuse_a=*/false, /*reuse_b=*/false);
        }

        // D rows are all equal (broadcast A): acc[0] = dot for column n.
        float dd = -acc[0];                       // d = -lsp
        dd = (dd <= 1.0f) ? (1.0f + 1e-6f) : dd;  // clamp below cosh(0)
        float arc = -logf(dd + sqrtf(dd * dd - 1.0f));   // -arcosh(d)

        d0 = (t == 0 && lane == 0) ? arc : d0;
        // each n appears in both lane halves -> count low half only
        sumExp += (valid && lowHalf) ? expf(arc) : 0.0f;
    }

    // ---- log-sum-exp butterfly over the wave ------------------------------
#pragma unroll
    for (int off = 16; off >= 1; off >>= 1)
        sumExp += __shfl_xor(sumExp, off, 32);

    if (lane == 0)
        out[b] = -(d0 - logf(sumExp + 1e-6f));
}

extern "C" void kernel_launch(void* const* d_in, const int* in_sizes, int n_in,
                              void* d_out, int out_size, void* d_ws, size_t ws_size,
                              hipStream_t stream) {
    const float* table = (const float*)d_in[0];
    const int*   I     = (const int*)d_in[1];
    const int*   Ks    = (const int*)d_in[2];
    float*       out   = (float*)d_out;

    const int Bcount = in_sizes[1];                         // 16384 anchors
    const int blocks = (Bcount + WAVES_PER_BLOCK - 1) / WAVES_PER_BLOCK;

    lorentz_loss_kernel<<<blocks, 32 * WAVES_PER_BLOCK, 0, stream>>>(
        table, I, Ks, out, Bcount);
}